// LiMNet_42838003810565
// MI455X (gfx1250) — compile-verified
//
#include <hip/hip_runtime.h>
#include <hip/hip_bf16.h>
#include <math.h>

// Problem constants (from reference)
#define B_    128
#define NU_   5000
#define NI_   5000
#define E_    128
#define UF_   16
#define IF_   16
#define D_    288   // 2E + UF + IF
#define G3E_  384   // 3E

typedef float v2f __attribute__((ext_vector_type(2)));
typedef float v8f __attribute__((ext_vector_type(8)));

// d_ws layout (floats):
//   [0,           B*D)        u_in   (128 x 288)
//   [B*D,       2*B*D)        i_in   (128 x 288)
//   [2*B*D,     2*B*D+B*3E)   gi_u   (128 x 384)
//   [.. ,       +B*3E)        gi_i   (128 x 384)
// total = 172,032 floats = 688,128 bytes

// ---------------------------------------------------------------------------
// 1) Gather: build u_in; i_in is u_in rotated by D/2 = 144 columns.
// ---------------------------------------------------------------------------
__global__ void limnet_gather(const int* __restrict__ uid,
                              const int* __restrict__ iid,
                              const float* __restrict__ ufeat,
                              const float* __restrict__ ifeat,
                              const float* __restrict__ umem,
                              const float* __restrict__ imem,
                              float* __restrict__ ws) {
  int t = blockIdx.x * blockDim.x + threadIdx.x;
  if (t >= B_ * D_) return;
  int b = t / D_;
  int k = t % D_;
  float v;
  if (k < E_) {
    v = umem[(size_t)b * NU_ * E_ + (size_t)uid[b] * E_ + k];
  } else if (k < E_ + UF_) {
    v = ufeat[b * UF_ + (k - E_)];
  } else if (k < 2 * E_ + UF_) {
    v = imem[(size_t)b * NI_ * E_ + (size_t)iid[b] * E_ + (k - E_ - UF_)];
  } else {
    v = ifeat[b * IF_ + (k - 2 * E_ - UF_)];
  }
  float* u_in = ws;
  float* i_in = ws + B_ * D_;
  u_in[b * D_ + k] = v;
  i_in[b * D_ + ((k + D_ / 2) % D_)] = v;   // i_in = rotate(u_in, 144)
}

// ---------------------------------------------------------------------------
// 2) GEMM via V_WMMA_F32_16X16X4_F32: gi = X @ W^T + bih
//    One wave per 16x16 output tile. 2 gemms * (128/16)*(384/16) = 384 waves.
//    A lane L: A[m0 + L%16][kk + (L/16)*2 + {0,1}]         (float2)
//    B lane L: W[n0 + L%16][kk + (L/16)*2 + {0,1}]  (B = W^T, so read W rows)
//    D lane L, vgpr v: D[m0 + (L/16)*8 + v][n0 + L%16]
// ---------------------------------------------------------------------------
__global__ void limnet_gemm_wmma(const float* __restrict__ Wu,
                                 const float* __restrict__ bih_u,
                                 const float* __restrict__ Wi,
                                 const float* __restrict__ bih_i,
                                 float* __restrict__ ws) {
  const int wave = blockIdx.x;          // 0..383 (blockDim.x == 32 -> one wave)
  const int g    = wave / 192;          // 0 = user gemm, 1 = item gemm
  const int tile = wave % 192;
  const int mt   = tile / 24;           // 8 row tiles
  const int nt   = tile % 24;           // 24 col tiles
  const int lane = threadIdx.x;         // 0..31 (wave32)
  const int r    = lane & 15;
  const int pair = (lane >> 4) << 1;    // 0 or 2

  const float* A   = ws + (g ? B_ * D_ : 0);
  const float* W   = g ? Wi : Wu;
  const float* bih = g ? bih_i : bih_u;
  float*       gi  = ws + 2 * B_ * D_ + (g ? B_ * G3E_ : 0);

  const float* arow = A + (size_t)(mt * 16 + r) * D_ + pair;
  const float* wrow = W + (size_t)(nt * 16 + r) * D_ + pair;

  v8f acc = {};
  #pragma unroll 8
  for (int kk = 0; kk < D_; kk += 4) {
    v2f a = *(const v2f*)(arow + kk);   // global_load_b64 (8B aligned)
    v2f b = *(const v2f*)(wrow + kk);
    acc = __builtin_amdgcn_wmma_f32_16x16x4_f32(
        /*neg_a=*/false, a, /*neg_b=*/false, b,
        /*c_mod=*/(short)0, acc, /*reuse_a=*/false, /*reuse_b=*/false);
  }

  const int   n    = nt * 16 + r;       // this lane's output column
  const float bias = bih[n];
  const int   m0   = mt * 16 + ((lane >> 4) << 3);
  #pragma unroll
  for (int v = 0; v < 8; ++v)
    gi[(size_t)(m0 + v) * G3E_ + n] = acc[v] + bias;
}

// ---------------------------------------------------------------------------
// 3) Epilogue: GRU gates (zero hidden) + L2 norm + write out + scatter into
//    the memory copies already placed in d_out. One wave32 per (gemm, row).
// ---------------------------------------------------------------------------
__global__ void limnet_epilogue(const int* __restrict__ uid,
                                const int* __restrict__ iid,
                                const float* __restrict__ bhh_u,
                                const float* __restrict__ bhh_i,
                                const float* __restrict__ ws,
                                float* __restrict__ out) {
  int gt   = blockIdx.x * blockDim.x + threadIdx.x;
  int wid  = gt >> 5;                   // 0..255
  int lane = threadIdx.x & 31;
  int g    = wid >> 7;                  // 0 = user, 1 = item
  int b    = wid & (B_ - 1);

  const float* gi  = ws + 2 * B_ * D_ + (g ? B_ * G3E_ : 0) + (size_t)b * G3E_;
  const float* bhh = g ? bhh_i : bhh_u;

  float h[4];
  float ss = 0.f;
  #pragma unroll
  for (int j = 0; j < 4; ++j) {
    int e = lane + 32 * j;              // 4 elements per lane, E = 128
    float rr = 1.f / (1.f + expf(-(gi[e] + bhh[e])));
    float zz = 1.f / (1.f + expf(-(gi[E_ + e] + bhh[E_ + e])));
    float nn = tanhf(gi[2 * E_ + e] + rr * bhh[2 * E_ + e]);
    h[j] = (1.f - zz) * nn;             // h' = (1-z)*n
    ss += h[j] * h[j];
  }
  // wave32 butterfly reduction for ||h||^2
  #pragma unroll
  for (int off = 16; off > 0; off >>= 1)
    ss += __shfl_xor(ss, off, 32);
  float inv = 1.f / fmaxf(sqrtf(ss), 1e-12f);

  const size_t memElems = (size_t)B_ * NU_ * E_;        // 81,920,000
  float* mem  = out + (size_t)B_ * 2 * E_ + (size_t)g * memElems;
  int    id   = g ? iid[b] : uid[b];
  float* mrow = mem + (size_t)b * NU_ * E_ + (size_t)id * E_;
  float* orow = out + (size_t)b * 2 * E_ + g * E_;

  #pragma unroll
  for (int j = 0; j < 4; ++j) {
    int e = lane + 32 * j;
    float hn = h[j] * inv;
    orow[e] = hn;   // out (B, 2E)
    mrow[e] = hn;   // scatter into updated memory copy
  }
}

// ---------------------------------------------------------------------------
extern "C" void kernel_launch(void* const* d_in, const int* in_sizes, int n_in,
                              void* d_out, int out_size, void* d_ws, size_t ws_size,
                              hipStream_t stream) {
  const int*   uid   = (const int*)d_in[0];
  const int*   iid   = (const int*)d_in[1];
  const float* ufeat = (const float*)d_in[2];
  const float* ifeat = (const float*)d_in[3];
  const float* umem  = (const float*)d_in[4];
  const float* imem  = (const float*)d_in[5];
  const float* Wu    = (const float*)d_in[6];
  const float* bihu  = (const float*)d_in[7];
  const float* bhhu  = (const float*)d_in[8];
  const float* Wi    = (const float*)d_in[9];
  const float* bihi  = (const float*)d_in[10];
  const float* bhhi  = (const float*)d_in[11];
  float* out = (float*)d_out;
  float* ws  = (float*)d_ws;

  const size_t memElems = (size_t)B_ * NU_ * E_;        // 81,920,000 floats

  // Dominant cost: return full updated memories (~655 MB out, ~56 us at
  // 23.3 TB/s). DMA d2d copies, then scatter the 128 updated rows on top.
  hipMemcpyAsync(out + (size_t)B_ * 2 * E_, umem,
                 memElems * sizeof(float), hipMemcpyDeviceToDevice, stream);
  hipMemcpyAsync(out + (size_t)B_ * 2 * E_ + memElems, imem,
                 memElems * sizeof(float), hipMemcpyDeviceToDevice, stream);

  limnet_gather<<<(B_ * D_ + 255) / 256, 256, 0, stream>>>(
      uid, iid, ufeat, ifeat, umem, imem, ws);
  limnet_gemm_wmma<<<384, 32, 0, stream>>>(Wu, bihu, Wi, bihi, ws);
  limnet_epilogue<<<64, 128, 0, stream>>>(uid, iid, bhhu, bhhi, ws, out);
}